// Encoder_23828478558447
// MI455X (gfx1250) — compile-verified
//
#include <hip/hip_runtime.h>
#include <hip/hip_bf16.h>
#include <math.h>

// ---------------------------------------------------------------------------
// Types for CDNA5 WMMA (f32 16x16x4, exact fp32 matmul on the matrix pipe)
// ---------------------------------------------------------------------------
typedef float v2f __attribute__((ext_vector_type(2)));
typedef float v8f __attribute__((ext_vector_type(8)));

// ---------------------------------------------------------------------------
// Small utility kernels
// ---------------------------------------------------------------------------
__global__ void fill_f32(float* __restrict__ p, float v, long long n) {
    long long i = (long long)blockIdx.x * blockDim.x + threadIdx.x;
    if (i < n) p[i] = v;
}

__global__ void copy_f32(float* __restrict__ dst, const float* __restrict__ src, long long n) {
    long long i = (long long)blockIdx.x * blockDim.x + threadIdx.x;
    if (i < n) dst[i] = src[i];
}

// in-place: p[i] = elu(p[i] + b[i & (F-1)])   (F is a power of two: 128 or 64)
__global__ void bias_elu(float* __restrict__ p, const float* __restrict__ b,
                         long long n, int fmask) {
    long long i = (long long)blockIdx.x * blockDim.x + threadIdx.x;
    if (i >= n) return;
    float x = p[i] + b[(int)(i & fmask)];
    p[i] = (x > 0.0f) ? x : expm1f(x);
}

// ---------------------------------------------------------------------------
// WMMA GEMM: C[M,Ncol] = A[M,K] * B[K,Ncol], fp32, one 16x16 tile per wave.
// K and Ncol are compile-time constants so every A/B load folds its K-offset
// into the 24-bit immediate IOFFSET field (no per-iteration 64-bit address
// arithmetic) and the 32-step WMMA chain fully unrolls.
// A-matrix VGPR layout (ISA 7.12.2, 32-bit A 16x4): lane(0..15)=row, laneHi
// selects K pair {0,1} vs {2,3}. B mirrored. C/D: vgpr v -> M = v + 8*laneHi,
// N = laneLo.
// ---------------------------------------------------------------------------
template <int K, int NCOL>
__global__ void gemm_wmma_f32(const float* __restrict__ A, const float* __restrict__ B,
                              float* __restrict__ C, int M) {
    const int wavesPerBlock = blockDim.x >> 5;
    const int tile    = blockIdx.x * wavesPerBlock + (threadIdx.x >> 5);
    const int tilesN  = NCOL >> 4;
    const int total   = (M >> 4) * tilesN;
    if (tile >= total) return;                 // wave-uniform; EXEC stays all-1s
    const int tm = tile / tilesN;
    const int tn = tile - tm * tilesN;
    const int lane   = threadIdx.x & 31;
    const int laneLo = lane & 15;
    const int laneHi = lane >> 4;

    const float* __restrict__ Arow = A + (size_t)(tm * 16 + laneLo) * K + 2 * laneHi;
    const float* __restrict__ Bcol = B + (size_t)(2 * laneHi) * NCOL + tn * 16 + laneLo;

    v8f acc = {};
#pragma unroll
    for (int k = 0; k < K; k += 4) {
        v2f a, b;
        a.x = Arow[k];
        a.y = Arow[k + 1];
        b.x = Bcol[k * NCOL];
        b.y = Bcol[(k + 1) * NCOL];
        acc = __builtin_amdgcn_wmma_f32_16x16x4_f32(
            /*neg_a=*/false, a, /*neg_b=*/false, b,
            /*c_mod=*/(short)0, acc, /*reuse_a=*/false, /*reuse_b=*/false);
    }

    float* __restrict__ Crow = C + (size_t)(tm * 16 + laneHi * 8) * NCOL + tn * 16 + laneLo;
#pragma unroll
    for (int v = 0; v < 8; ++v) Crow[v * NCOL] = acc[v];
}

// ---------------------------------------------------------------------------
// Per-(node, head) attention coefficients: al_s/al_d = <xp[n,h,:], a_src/a_dst[h,:]>
// ---------------------------------------------------------------------------
__global__ void attn_coef(const float* __restrict__ xp,
                          const float* __restrict__ a_src, const float* __restrict__ a_dst,
                          float* __restrict__ als, float* __restrict__ ald,
                          int Nn, int C) {
    int idx = blockIdx.x * blockDim.x + threadIdx.x;  // idx = n*2 + h
    if (idx >= Nn * 2) return;
    int n = idx >> 1, h = idx & 1;
    const float* row = xp + (size_t)n * 2 * C + (size_t)h * C;
    float ss = 0.f, sd = 0.f;
    for (int c = 0; c < C; ++c) {
        float v = row[c];
        ss += v * a_src[h * C + c];
        sd += v * a_dst[h * C + c];
    }
    als[idx] = ss;
    ald[idx] = sd;
}

// ---------------------------------------------------------------------------
// Edge helpers
// ---------------------------------------------------------------------------
__device__ __forceinline__ void edge_src_dst(const int* __restrict__ ei, long long e,
                                             long long E, int& src, int& dst) {
    if (e < E) { src = ei[e]; dst = ei[E + e]; }
    else       { src = dst = (int)(e - E); }           // self loops appended
}

__device__ __forceinline__ float lrelu02(float x) {
    return (x > 0.0f) ? x : 0.2f * x;
}

// float atomic max via monotonic int/uint punning (init value must be -inf)
__device__ __forceinline__ void atomicMaxF(float* addr, float val) {
    if (val >= 0.0f) atomicMax((int*)addr, __float_as_int(val));
    else             atomicMin((unsigned int*)addr, __float_as_uint(val));
}

// pass 1: segment max of leaky-relu logits per (dst, head)
__global__ void edge_logit_max(const int* __restrict__ ei, long long E, int Nn,
                               const float* __restrict__ als, const float* __restrict__ ald,
                               float* __restrict__ m) {
    long long e = (long long)blockIdx.x * blockDim.x + threadIdx.x;
    if (e >= E + Nn) return;
    int src, dst;
    edge_src_dst(ei, e, E, src, dst);
#pragma unroll
    for (int h = 0; h < 2; ++h) {
        float l = lrelu02(als[src * 2 + h] + ald[dst * 2 + h]);
        atomicMaxF(&m[dst * 2 + h], l);
    }
}

// pass 2: segment sum of exp(logit - m[dst]) per (dst, head)
__global__ void edge_expsum(const int* __restrict__ ei, long long E, int Nn,
                            const float* __restrict__ als, const float* __restrict__ ald,
                            const float* __restrict__ m, float* __restrict__ den) {
    long long e = (long long)blockIdx.x * blockDim.x + threadIdx.x;
    if (e >= E + Nn) return;
    int src, dst;
    edge_src_dst(ei, e, E, src, dst);
#pragma unroll
    for (int h = 0; h < 2; ++h) {
        float l = lrelu02(als[src * 2 + h] + ald[dst * 2 + h]);
        atomicAdd(&den[dst * 2 + h], expf(l - m[dst * 2 + h]));
    }
}

// pass 3: agg[dst] += alpha * xp[src].  One wave per edge; lane owns CPL
// contiguous channels (all inside one head since CPL*16 == channels per head).
// F = 32*CPL total feature channels (128 -> CPL=4, 64 -> CPL=2).
template <int CPL>
__global__ void edge_aggregate(const int* __restrict__ ei, long long E, int Nn,
                               const float* __restrict__ als, const float* __restrict__ ald,
                               const float* __restrict__ m, const float* __restrict__ den,
                               const float* __restrict__ xp, float* __restrict__ agg) {
    const int F = 32 * CPL;
    const int wavesPerBlock = blockDim.x >> 5;
    long long e = (long long)blockIdx.x * wavesPerBlock + (threadIdx.x >> 5);
    if (e >= E + Nn) return;
    const int lane = threadIdx.x & 31;
    int src, dst;
    edge_src_dst(ei, e, E, src, dst);

    const int c0 = lane * CPL;
    const int h  = (c0 >= (F >> 1)) ? 1 : 0;
    float l = lrelu02(als[src * 2 + h] + ald[dst * 2 + h]);
    float alpha = expf(l - m[dst * 2 + h]) / den[dst * 2 + h];

    const float* __restrict__ xr = xp + (size_t)src * F + c0;
    float*       __restrict__ ar = agg + (size_t)dst * F + c0;
#pragma unroll
    for (int j = 0; j < CPL; ++j) atomicAdd(&ar[j], xr[j] * alpha);
}

// ---------------------------------------------------------------------------
// Host-side launch
// ---------------------------------------------------------------------------
static inline int cdiv(long long a, int b) { return (int)((a + b - 1) / b); }

extern "C" void kernel_launch(void* const* d_in, const int* in_sizes, int n_in,
                              void* d_out, int out_size, void* d_ws, size_t ws_size,
                              hipStream_t stream) {
    const float* x      = (const float*)d_in[0];
    const int*   ei     = (const int*)  d_in[1];
    const float* W1     = (const float*)d_in[2];
    const float* a_src1 = (const float*)d_in[3];
    const float* a_dst1 = (const float*)d_in[4];
    const float* b1     = (const float*)d_in[5];
    const float* W2     = (const float*)d_in[6];
    const float* a_src2 = (const float*)d_in[7];
    const float* a_dst2 = (const float*)d_in[8];
    const float* b2     = (const float*)d_in[9];

    const int       Nn   = in_sizes[0] / 128;   // 50000
    const long long E    = in_sizes[1] / 2;     // 1,600,000
    const long long Etot = E + Nn;              // + self loops

    float* out  = (float*)d_out;                // layer-2 result [Nn,64]
    float* enc  = out + (size_t)Nn * 64;        // encoded passthrough [Nn,128]

    // workspace layout (floats)
    float* ws   = (float*)d_ws;
    float* xp1  = ws;                            // Nn*128
    float* agg1 = xp1  + (size_t)Nn * 128;       // Nn*128 (becomes h after bias+elu)
    float* xp2  = agg1 + (size_t)Nn * 128;       // Nn*64
    float* al1s = xp2  + (size_t)Nn * 64;        // Nn*2 each below
    float* al1d = al1s + (size_t)Nn * 2;
    float* m1   = al1d + (size_t)Nn * 2;
    float* den1 = m1   + (size_t)Nn * 2;
    float* al2s = den1 + (size_t)Nn * 2;
    float* al2d = al2s + (size_t)Nn * 2;
    float* m2   = al2d + (size_t)Nn * 2;
    float* den2 = m2   + (size_t)Nn * 2;

    const int BT = 256;                          // 8 wave32s per block

    // ---- init (every call: determinism; harness does not re-poison) ----
    fill_f32<<<cdiv((long long)Nn * 128, BT), BT, 0, stream>>>(agg1, 0.0f, (long long)Nn * 128);
    fill_f32<<<cdiv((long long)Nn * 64,  BT), BT, 0, stream>>>(out,  0.0f, (long long)Nn * 64);
    fill_f32<<<cdiv((long long)Nn * 2,   BT), BT, 0, stream>>>(m1,  -INFINITY, (long long)Nn * 2);
    fill_f32<<<cdiv((long long)Nn * 2,   BT), BT, 0, stream>>>(den1, 0.0f, (long long)Nn * 2);
    fill_f32<<<cdiv((long long)Nn * 2,   BT), BT, 0, stream>>>(m2,  -INFINITY, (long long)Nn * 2);
    fill_f32<<<cdiv((long long)Nn * 2,   BT), BT, 0, stream>>>(den2, 0.0f, (long long)Nn * 2);

    // ---- layer 1 ----
    {
        const int tiles = (Nn >> 4) * (128 >> 4);
        gemm_wmma_f32<128, 128><<<cdiv(tiles, 8), BT, 0, stream>>>(x, W1, xp1, Nn);
        attn_coef<<<cdiv((long long)Nn * 2, BT), BT, 0, stream>>>(xp1, a_src1, a_dst1, al1s, al1d, Nn, 64);
        edge_logit_max<<<cdiv(Etot, BT), BT, 0, stream>>>(ei, E, Nn, al1s, al1d, m1);
        edge_expsum  <<<cdiv(Etot, BT), BT, 0, stream>>>(ei, E, Nn, al1s, al1d, m1, den1);
        edge_aggregate<4><<<cdiv(Etot, 8), BT, 0, stream>>>(ei, E, Nn, al1s, al1d, m1, den1, xp1, agg1);
        bias_elu<<<cdiv((long long)Nn * 128, BT), BT, 0, stream>>>(agg1, b1, (long long)Nn * 128, 127);
        // agg1 is now h [Nn,128]
    }

    // ---- layer 2 ----
    {
        const int tiles = (Nn >> 4) * (64 >> 4);
        gemm_wmma_f32<128, 64><<<cdiv(tiles, 8), BT, 0, stream>>>(agg1, W2, xp2, Nn);
        attn_coef<<<cdiv((long long)Nn * 2, BT), BT, 0, stream>>>(xp2, a_src2, a_dst2, al2s, al2d, Nn, 32);
        edge_logit_max<<<cdiv(Etot, BT), BT, 0, stream>>>(ei, E, Nn, al2s, al2d, m2);
        edge_expsum  <<<cdiv(Etot, BT), BT, 0, stream>>>(ei, E, Nn, al2s, al2d, m2, den2);
        edge_aggregate<2><<<cdiv(Etot, 8), BT, 0, stream>>>(ei, E, Nn, al2s, al2d, m2, den2, xp2, out);
        bias_elu<<<cdiv((long long)Nn * 64, BT), BT, 0, stream>>>(out, b2, (long long)Nn * 64, 63);
    }

    // ---- encoded passthrough: second tuple element is x itself ----
    copy_f32<<<cdiv((long long)Nn * 128, BT), BT, 0, stream>>>(enc, x, (long long)Nn * 128);
}